// ScannedRNN_16810501996978
// MI455X (gfx1250) — compile-verified
//
#include <hip/hip_runtime.h>

// ---------------- GRU scan for CDNA5 (gfx1250), fp32 WMMA 16x16x4 ----------------
// T=1024, B=256, D=256. 16 workgroups x 512 threads (16 waves).
// Each workgroup owns 16 batch rows for all timesteps (batch rows are
// independent in the recurrence -> no cross-WG sync). Wave w computes the
// six 16x16 tiles {xWi_r, xWi_z, xWi_n, hWh_r, hWh_z, hWh_n} for h-columns
// [16w,16w+16), accumulating over K=256 with v_wmma_f32_16x16x4_f32,
// then does the gate math in registers.

typedef __attribute__((ext_vector_type(2))) float v2f;
typedef __attribute__((ext_vector_type(8))) float v8f;

#define T_STEPS 1024
#define BATCH   256
#define DIM     256
#define G3      768          // 3*DIM gate columns
#define KT      64           // 256 / 4 k-slices
#define HST     260          // padded LDS row pitch (floats): bank-conflict-free b64 A loads
#define NWAVE   16
#define BLK     (NWAVE * 32)
#define NTILE_PER_MAT 48     // 768 / 16
#define PACK_ELEMS (2 * NTILE_PER_MAT * KT * 64)   // 393216 floats = 1.5 MB

// --- one-time weight repack into WMMA B-operand order --------------------------
// dest[((m*48+j)*64+kk)*64 + lane*2 + v] = W_m[K=4kk+2*(lane/16)+v][N=16j+lane%16]
// m=0: Wi (D x 3D). m=1: [Whr|Whz|Whn] concatenated along N.
__global__ __launch_bounds__(256) void gru_pack_weights(
    const float* __restrict__ Wi,  const float* __restrict__ Whr,
    const float* __restrict__ Whz, const float* __restrict__ Whn,
    float* __restrict__ wpack) {
  int idx = blockIdx.x * 256 + threadIdx.x;
  if (idx >= PACK_ELEMS) return;
  int lv   = idx & 63;
  int lane = lv >> 1;
  int v    = lv & 1;
  int kk   = (idx >> 6) & 63;
  int mj   = idx >> 12;                 // 0..95
  int m    = (mj >= NTILE_PER_MAT) ? 1 : 0;
  int j    = mj - m * NTILE_PER_MAT;
  int K    = 4 * kk + 2 * (lane >> 4) + v;
  int N    = 16 * j + (lane & 15);
  float val;
  if (m == 0) {
    val = Wi[K * G3 + N];
  } else {
    if      (N < 256) val = Whr[K * DIM + N];
    else if (N < 512) val = Whz[K * DIM + (N - 256)];
    else              val = Whn[K * DIM + (N - 512)];
  }
  wpack[idx] = val;
}

// --- the scan ------------------------------------------------------------------
__global__ __launch_bounds__(BLK) void gru_scan(
    const float* __restrict__ x,        // (T,B,D)
    const unsigned char* __restrict__ resets,  // (T,B) bool
    const float* __restrict__ bi,       // (3D)
    const float* __restrict__ bhn,      // (D)
    const float* __restrict__ wpack,    // packed B operands
    float* __restrict__ out) {          // (T,B,D)
  __shared__ float hbuf[16 * HST];
  __shared__ float xbuf[16 * HST];
  __shared__ float bis[G3];
  __shared__ float bhns[DIM];

  const int tid  = threadIdx.x;
  const int lane = tid & 31;
  const int w    = tid >> 5;            // wave id 0..15
  const int b0   = blockIdx.x * 16;     // this WG's batch rows

  for (int e = tid; e < 16 * HST; e += BLK) hbuf[e] = 0.0f;   // h0 = 0
  for (int e = tid; e < G3;  e += BLK) bis[e]  = bi[e];
  for (int e = tid; e < DIM; e += BLK) bhns[e] = bhn[e];
  __syncthreads();

  const int mrow  = lane & 15;          // A-operand row (batch row in tile)
  const int khalf = (lane >> 4) << 1;   // A/B k sub-offset for upper half-wave

  // packed B base pointers for this wave's six tiles
  const float* wbase[6];
  wbase[0] = wpack + ((size_t)(0 * NTILE_PER_MAT + w     ) * KT) * 64 + lane * 2;
  wbase[1] = wpack + ((size_t)(0 * NTILE_PER_MAT + w + 16) * KT) * 64 + lane * 2;
  wbase[2] = wpack + ((size_t)(0 * NTILE_PER_MAT + w + 32) * KT) * 64 + lane * 2;
  wbase[3] = wpack + ((size_t)(1 * NTILE_PER_MAT + w     ) * KT) * 64 + lane * 2;
  wbase[4] = wpack + ((size_t)(1 * NTILE_PER_MAT + w + 16) * KT) * 64 + lane * 2;
  wbase[5] = wpack + ((size_t)(1 * NTILE_PER_MAT + w + 32) * KT) * 64 + lane * 2;

  const int col   = (w << 4) + (lane & 15);   // this lane's h column
  const int rbase = (lane >> 4) << 3;         // D-tile row base: 0 or 8
  const float brc = bis[col], bzc = bis[col + 256], bnc = bis[col + 512];
  const float bhnc = bhns[col];

  for (int t = 0; t < T_STEPS; ++t) {
    // ---- phase 1: episode resets + stage x_t tile into LDS (coalesced) ----
    const float* xt = x + (size_t)t * BATCH * DIM + (size_t)b0 * DIM;
    for (int e = tid; e < 16 * DIM; e += BLK) {
      int r = e >> 8, c = e & 255;
      xbuf[r * HST + c] = xt[r * DIM + c];
      if (resets[(size_t)t * BATCH + b0 + r]) hbuf[r * HST + c] = 0.0f;
    }
    __syncthreads();

    // ---- phase 2: six 16x16x256 WMMA tile-GEMMs, fully in registers ----
    v8f acc[6];
#pragma unroll
    for (int i = 0; i < 6; ++i) acc[i] = (v8f){0.f,0.f,0.f,0.f,0.f,0.f,0.f,0.f};

#pragma unroll 4
    for (int kk = 0; kk < KT; ++kk) {
      const int ko = 4 * kk + khalf;
      v2f ax = *(const v2f*)&xbuf[mrow * HST + ko];   // ds_load_b64, bank-conflict-free
      v2f ah = *(const v2f*)&hbuf[mrow * HST + ko];
#pragma unroll
      for (int i = 0; i < 6; ++i) {
        v2f b = *(const v2f*)(wbase[i] + kk * 64);    // coalesced global_load_b64 (L2-resident)
        v2f a = (i < 3) ? ax : ah;
        acc[i] = __builtin_amdgcn_wmma_f32_16x16x4_f32(
            /*neg_a=*/false, a, /*neg_b=*/false, b,
            /*c_mod=*/(short)0, acc[i], /*reuse_a=*/false, /*reuse_b=*/false);
      }
    }

    // prefetch next step's x tile into cache (global_prefetch_b8)
    if (t + 1 < T_STEPS && tid < 256) {
      __builtin_prefetch(xt + (size_t)BATCH * DIM + (size_t)tid * 16, 0, 0);
    }
    __syncthreads();   // all WMMA reads of hbuf done before we overwrite h

    // ---- phase 3: gate math in registers; update h and emit output ----
    float* outp = out + (size_t)t * BATCH * DIM + (size_t)b0 * DIM;
#pragma unroll
    for (int v = 0; v < 8; ++v) {
      const int row = rbase + v;                      // C/D layout: M = v + 8*(lane>=16)
      float hprev = hbuf[row * HST + col];
      float ar = acc[0][v] + brc, az = acc[1][v] + bzc, an = acc[2][v] + bnc;
      float gr = acc[3][v],       gz = acc[4][v],       gn = acc[5][v] + bhnc;
      float rg = 1.0f / (1.0f + expf(-(ar + gr)));
      float zg = 1.0f / (1.0f + expf(-(az + gz)));
      float ng = tanhf(an + rg * gn);
      float hn = (1.0f - zg) * ng + zg * hprev;
      hbuf[row * HST + col] = hn;
      outp[(size_t)row * DIM + col] = hn;
    }
    __syncthreads();   // h fully updated before next step's reset/x staging
  }
}

extern "C" void kernel_launch(void* const* d_in, const int* in_sizes, int n_in,
                              void* d_out, int out_size, void* d_ws, size_t ws_size,
                              hipStream_t stream) {
  const float*         x      = (const float*)d_in[0];
  const unsigned char* resets = (const unsigned char*)d_in[1];
  const float*         Wi     = (const float*)d_in[2];
  const float*         bi     = (const float*)d_in[3];
  const float*         Whr    = (const float*)d_in[4];
  const float*         Whz    = (const float*)d_in[5];
  const float*         Whn    = (const float*)d_in[6];
  const float*         bhn    = (const float*)d_in[7];
  float* out   = (float*)d_out;
  float* wpack = (float*)d_ws;          // needs 1.5 MB scratch

  gru_pack_weights<<<(PACK_ELEMS + 255) / 256, 256, 0, stream>>>(Wi, Whr, Whz, Whn, wpack);
  gru_scan<<<BATCH / 16, BLK, 0, stream>>>(x, resets, bi, bhn, wpack, out);
}